// Operator_36988258353306
// MI455X (gfx1250) — compile-verified
//
#include <hip/hip_runtime.h>

// ---------------- problem constants ----------------
constexpr int kH   = 720;
constexpr int kW   = 1440;
constexpr int kHW  = kH * kW;          // 1,036,800
constexpr int kT   = 10;
constexpr int kNM  = 9;                // T-1 modes
constexpr int kNCH = 6;
constexpr int kCLAT = 4;

// Gram-kernel partition: 270 blocks * 8 waves * 480 px = kHW exactly
constexpr int kNB   = 270;
constexpr int kWAVES = 8;
constexpr int kPixPerBlock = kHW / kNB;              // 3840
constexpr int kPixPerWave  = kPixPerBlock / kWAVES;  // 480 (multiple of 8)

// ---------------- workspace layout (floats) ----------------
constexpr size_t WS_FX    = 0;                       // 9*kHW
constexpr size_t WS_FY    = (size_t)9  * kHW;        // 9*kHW
constexpr size_t WS_ADVX9 = (size_t)18 * kHW;        // kHW
constexpr size_t WS_ADVY9 = (size_t)19 * kHW;        // kHW
constexpr size_t WS_INX   = (size_t)20 * kHW;        // 4*kHW
constexpr size_t WS_INY   = (size_t)24 * kHW;        // 4*kHW
constexpr size_t WS_PART  = (size_t)28 * kHW;        // 2*kNB*256
constexpr size_t WS_G     = WS_PART + (size_t)2 * kNB * 256; // 2*256
constexpr size_t WS_WVEC  = WS_G + 512;              // 2*16

typedef float v2f __attribute__((ext_vector_type(2)));
typedef float v8f __attribute__((ext_vector_type(8)));

// ---------------- grid constants (match jnp float32 math) ----------------
__device__ __forceinline__ float lat_rad(int h) {
  return (float)(6.283185307179586 / 719.0) * (float)h;   // pi*linspace(0,360,720)/180
}
__device__ __forceinline__ float dx_of(int h) {
  // (1/720)*2*pi*(6370*cos(lat))*1000/(6*3600)
  const float DXF = (float)(2.0 * 3.141592653589793 * 6.37e6 / 15552000.0);
  return DXF * __cosf(lat_rad(h));
}
__device__ __forceinline__ float f_of(int h) {
  return 1.46e-4f * __sinf(lat_rad(h));   // 2*7.3e-5*sin
}

__device__ __forceinline__ float ldx(const float* __restrict__ x, int t, int ch, int h, int w) {
  return x[((size_t)(t * kNCH + ch) * kH + h) * kW + w];
}

__device__ __forceinline__ float grad_lon_x(const float* __restrict__ x, int t, int ch,
                                            int h, int w, float dxh) {
  if (w == 0)        return (ldx(x,t,ch,h,1)      - ldx(x,t,ch,h,0))      / dxh;
  if (w == kW - 1)   return (ldx(x,t,ch,h,kW-1)   - ldx(x,t,ch,h,kW-2))   / dxh;
  return (ldx(x,t,ch,h,w+1) - ldx(x,t,ch,h,w-1)) / (2.0f * dxh);
}

__device__ __forceinline__ float grad_lat_x(const float* __restrict__ x, int t, int ch,
                                            int h, int w) {
  if (h == 0)        return (ldx(x,t,ch,1,w)      - ldx(x,t,ch,0,w))      / dx_of(1);
  if (h == kH - 1)   return (ldx(x,t,ch,kH-1,w)   - ldx(x,t,ch,kH-2,w))   / dx_of(kH-1);
  float hl = dx_of(h), hr = dx_of(h + 1);
  float fm = ldx(x,t,ch,h-1,w), f0 = ldx(x,t,ch,h,w), fp = ldx(x,t,ch,h+1,w);
  return (hl*hl*fp + (hr*hr - hl*hl)*f0 - hr*hr*fm) / (hl * hr * (hl + hr));
}

// ============ K1: advection terms; fx/fy fields; advx9/advy9 ============
__global__ void adv_kernel(const float* __restrict__ x, float* __restrict__ ws) {
  int p = blockIdx.x * blockDim.x + threadIdx.x;
  if (p >= kHW) return;
  int t = blockIdx.y;
  int h = p / kW, w = p % kW;
  float dxh = dx_of(h);
  float u0 = ldx(x,t,0,h,w), v0 = ldx(x,t,1,h,w);
  float ux = grad_lon_x(x,t,0,h,w,dxh), uy = grad_lat_x(x,t,0,h,w);
  float vx = grad_lon_x(x,t,1,h,w,dxh), vy = grad_lat_x(x,t,1,h,w);
  float fc = f_of(h);
  float advx = u0*ux + v0*uy + fc*v0;
  float advy = u0*vx + v0*vy - fc*u0;
  if (t < kT - 1) {
    ws[WS_FX + (size_t)t*kHW + p] = (ldx(x,t+1,0,h,w) - u0) + advx;
    ws[WS_FY + (size_t)t*kHW + p] = (ldx(x,t+1,1,h,w) - v0) + advy;
  } else {
    ws[WS_ADVX9 + p] = advx;
    ws[WS_ADVY9 + p] = advy;
  }
}

// ============ K2: Gram matrices via V_WMMA_F32_16X16X4_F32 ============
// G[t,t'] = sum_p f[t,p]*f[t',p]; 9x9 padded into 16x16, K=4 pixels/WMMA.
// A-matrix f32 16x4 lane layout (ISA 7.12.2): lane L -> M=L%16,
// VGPR0 holds K=(L<16?0:2), VGPR1 K=(L<16?1:3). B mirrors A, so one v2f
// feeds both operands for a symmetric Gram.
// Padding lanes (t>=9) read row 0 unconditionally (valid memory) and are
// zeroed by a 0/1 mask multiply -> unconditional b64 loads, no EXEC churn.
__global__ void gram_wmma(const float* __restrict__ ws, float* __restrict__ part) {
  const float* f = ws + (blockIdx.y ? WS_FY : WS_FX);
  int lane = threadIdx.x & 31;
  int wave = threadIdx.x >> 5;
  int t = lane & 15;
  float m = (t < kNM) ? 1.0f : 0.0f;
  const float* row = f + (size_t)((t < kNM) ? t : 0) * kHW;
  const v2f* row2 = (const v2f*)row;
  // base pixel for this lane: even -> v2f index = base>>1
  int base = blockIdx.x * kPixPerBlock + wave * kPixPerWave + ((lane >> 4) << 1);
  int idx = base >> 1;
  v8f acc0 = {}, acc1 = {};
  for (int p = 0; p < kPixPerWave; p += 8) {
    v2f a0 = row2[idx];       // pixels base+p, base+p+1
    v2f a1 = row2[idx + 2];   // pixels base+p+4, base+p+5
    a0.x *= m; a0.y *= m;
    a1.x *= m; a1.y *= m;
    acc0 = __builtin_amdgcn_wmma_f32_16x16x4_f32(false, a0, false, a0,
                                                 (short)0, acc0, false, false);
    acc1 = __builtin_amdgcn_wmma_f32_16x16x4_f32(false, a1, false, a1,
                                                 (short)0, acc1, false, false);
    idx += 4;
  }
  v8f acc = acc0 + acc1;
  // deterministic block reduction (fragment layout preserved: i = r*32+lane)
  __shared__ float red[kWAVES * 256];
#pragma unroll
  for (int r = 0; r < 8; ++r) red[wave * 256 + r * 32 + lane] = acc[r];
  __syncthreads();
  float s = 0.0f;
  int i = threadIdx.x;
#pragma unroll
  for (int wv = 0; wv < kWAVES; ++wv) s += red[wv * 256 + i];
  part[((size_t)blockIdx.y * kNB + blockIdx.x) * 256 + i] = s;
}

__global__ void gram_reduce(const float* __restrict__ part, float* __restrict__ Gout) {
  int fld = blockIdx.x, i = threadIdx.x;
  float s = 0.0f;
  for (int b = 0; b < kNB; ++b) s += part[((size_t)fld * kNB + b) * 256 + i];
  Gout[fld * 256 + i] = s;
}

// fragment-layout accessor: element (M,N) of 16x16 f32 C/D
__device__ __forceinline__ float fragG(const float* __restrict__ G, int M, int N) {
  return G[(M & 7) * 32 + (M >> 3) * 16 + N];
}

// ============ K3: 9x9 Jacobi eigensolve + mode-weight vector ============
// fxs[8] = sum_t w[t]*fx[t],  w = ((G[8,:]·Vd^T)·A)·Vd,  Vd[m,t]=d[m]*V[t,m],
// d = s/(s^2 + [s<=1]),  s = sqrt(eig).  One thread per field.
__global__ void modes_kernel(const float* __restrict__ ws_G,
                             const float* __restrict__ Ax,
                             const float* __restrict__ Ay,
                             float* __restrict__ ws_w) {
  int fld = threadIdx.x;
  if (fld >= 2) return;
  const float* G = ws_G + fld * 256;
  const float* A = fld ? Ay : Ax;
  float g[kNM][kNM], V[kNM][kNM];
  for (int i = 0; i < kNM; ++i)
    for (int j = 0; j < kNM; ++j) {
      g[i][j] = 0.5f * (fragG(G, i, j) + fragG(G, j, i));
      V[i][j] = (i == j) ? 1.0f : 0.0f;
    }
  for (int sweep = 0; sweep < 16; ++sweep)
    for (int p = 0; p < kNM - 1; ++p)
      for (int q = p + 1; q < kNM; ++q) {
        float apq = g[p][q];
        if (fabsf(apq) <= 1e-30f) continue;
        float th = (g[q][q] - g[p][p]) / (2.0f * apq);
        float tq = ((th >= 0.0f) ? 1.0f : -1.0f) / (fabsf(th) + sqrtf(th * th + 1.0f));
        float c = 1.0f / sqrtf(tq * tq + 1.0f);
        float s = tq * c;
        for (int k = 0; k < kNM; ++k) {
          float gkp = g[k][p], gkq = g[k][q];
          g[k][p] = c * gkp - s * gkq; g[k][q] = s * gkp + c * gkq;
        }
        for (int k = 0; k < kNM; ++k) {
          float gpk = g[p][k], gqk = g[q][k];
          g[p][k] = c * gpk - s * gqk; g[q][k] = s * gpk + c * gqk;
        }
        for (int k = 0; k < kNM; ++k) {
          float vkp = V[k][p], vkq = V[k][q];
          V[k][p] = c * vkp - s * vkq; V[k][q] = s * vkp + c * vkq;
        }
      }
  float Vd[kNM][kNM];
  for (int m = 0; m < kNM; ++m) {
    float s = sqrtf(fmaxf(g[m][m], 0.0f));
    float a2 = (s > 1.0f) ? 0.0f : 1.0f;
    float d = s / (s * s + a2);
    for (int t = 0; t < kNM; ++t) Vd[m][t] = d * V[t][m];
  }
  float coef8[kNM], g2[kNM];
  for (int m = 0; m < kNM; ++m) {
    float s = 0.0f;
    for (int t = 0; t < kNM; ++t)
      s += Vd[m][t] * 0.5f * (fragG(G, t, kNM - 1) + fragG(G, kNM - 1, t));
    coef8[m] = s;
  }
  for (int m = 0; m < kNM; ++m) {
    float s = 0.0f;
    for (int M = 0; M < kNM; ++M) s += coef8[M] * A[M * kNM + m];
    g2[m] = s;
  }
  for (int t = 0; t < kNM; ++t) {
    float s = 0.0f;
    for (int m = 0; m < kNM; ++m) s += g2[m] * Vd[m][t];
    ws_w[fld * 16 + t] = s;
  }
}

// ============ K4: u2/v2 = u[9] + sum_t w[t]*f[t] - adv[9] ============
__global__ void final_uv(const float* __restrict__ x, const float* __restrict__ ws,
                         float* __restrict__ out) {
  int p = blockIdx.x * blockDim.x + threadIdx.x;
  if (p >= kHW) return;
  const float* wx = ws + WS_WVEC;
  const float* wy = ws + WS_WVEC + 16;
  float su = 0.0f, sv = 0.0f;
#pragma unroll
  for (int t = 0; t < kNM; ++t) {
    su += wx[t] * ws[WS_FX + (size_t)t * kHW + p];
    sv += wy[t] * ws[WS_FY + (size_t)t * kHW + p];
  }
  float u9 = x[((size_t)(9 * kNCH + 0)) * kHW + p];
  float v9 = x[((size_t)(9 * kNCH + 1)) * kHW + p];
  out[p]        = u9 + su - ws[WS_ADVX9 + p];
  out[kHW + p]  = v9 + sv - ws[WS_ADVY9 + p];
}

// ============ K5a: innerx/innery at t=9 ============
__global__ void inner_kernel(const float* __restrict__ x, const float* __restrict__ rate,
                             const float* __restrict__ dif, float* __restrict__ ws) {
  int p = blockIdx.x * blockDim.x + threadIdx.x;
  if (p >= kHW) return;
  int j = blockIdx.y;
  int h = p / kW, w = p % kW;
  int ch = 2 + j;
  float dxh = dx_of(h);
  float c9 = ldx(x, 9, ch, h, w);
  float cx = grad_lon_x(x, 9, ch, h, w, dxh);
  float cy = grad_lat_x(x, 9, ch, h, w);
  float u9 = ldx(x, 9, 0, h, w), v9 = ldx(x, 9, 1, h, w);
  size_t rbase = ((size_t)(j * kH + h) * kW + w) * 2;
  float dj = dif[j];
  ws[WS_INX + (size_t)j * kHW + p] = cx * dj - rate[rbase + 0] * u9 * c9;
  ws[WS_INY + (size_t)j * kHW + p] = cy * dj - rate[rbase + 1] * v9 * c9;
}

// ============ K5b: c2 = c[9] + grad_lon(innerx) + grad_lat(innery) ============
__global__ void final_c(const float* __restrict__ x, const float* __restrict__ ws,
                        float* __restrict__ out) {
  int p = blockIdx.x * blockDim.x + threadIdx.x;
  if (p >= kHW) return;
  int j = blockIdx.y;
  int h = p / kW, w = p % kW;
  const float* inx = ws + WS_INX + (size_t)j * kHW;
  const float* iny = ws + WS_INY + (size_t)j * kHW;
  float dxh = dx_of(h);
  float gl;
  if (w == 0)           gl = (inx[(size_t)h*kW + 1]    - inx[(size_t)h*kW])        / dxh;
  else if (w == kW - 1) gl = (inx[(size_t)h*kW + kW-1] - inx[(size_t)h*kW + kW-2]) / dxh;
  else                  gl = (inx[(size_t)h*kW + w+1]  - inx[(size_t)h*kW + w-1])  / (2.0f*dxh);
  float gt;
  if (h == 0)           gt = (iny[(size_t)kW + w] - iny[w]) / dx_of(1);
  else if (h == kH - 1) gt = (iny[(size_t)(kH-1)*kW + w] - iny[(size_t)(kH-2)*kW + w]) / dx_of(kH-1);
  else {
    float hl = dx_of(h), hr = dx_of(h + 1);
    float fm = iny[(size_t)(h-1)*kW + w], f0 = iny[(size_t)h*kW + w], fp = iny[(size_t)(h+1)*kW + w];
    gt = (hl*hl*fp + (hr*hr - hl*hl)*f0 - hr*hr*fm) / (hl * hr * (hl + hr));
  }
  float c9 = ldx(x, 9, 2 + j, h, w);
  out[(size_t)(2 + j) * kHW + p] = c9 + gl + gt;
}

extern "C" void kernel_launch(void* const* d_in, const int* in_sizes, int n_in,
                              void* d_out, int out_size, void* d_ws, size_t ws_size,
                              hipStream_t stream) {
  const float* x    = (const float*)d_in[0];
  const float* rate = (const float*)d_in[1];
  const float* dif  = (const float*)d_in[2];
  const float* Ax   = (const float*)d_in[3];
  const float* Ay   = (const float*)d_in[4];
  float* out = (float*)d_out;
  float* ws  = (float*)d_ws;

  dim3 blk(256);
  adv_kernel  <<<dim3(kHW / 256, kT), blk, 0, stream>>>(x, ws);
  gram_wmma   <<<dim3(kNB, 2),        blk, 0, stream>>>(ws, ws + WS_PART);
  gram_reduce <<<dim3(2),             blk, 0, stream>>>(ws + WS_PART, ws + WS_G);
  modes_kernel<<<dim3(1), dim3(32),        0, stream>>>(ws + WS_G, Ax, Ay, ws + WS_WVEC);
  final_uv    <<<dim3(kHW / 256),     blk, 0, stream>>>(x, ws, out);
  inner_kernel<<<dim3(kHW / 256, kCLAT), blk, 0, stream>>>(x, rate, dif, ws);
  final_c     <<<dim3(kHW / 256, kCLAT), blk, 0, stream>>>(x, ws, out);
}